// Banyan_81793357185445
// MI455X (gfx1250) — compile-verified
//
#include <hip/hip_runtime.h>
#include <hip/hip_bf16.h>
#include <math.h>

// ---------------------------------------------------------------------------
// Banyan greedy adjacent-pair merge, MI455X (gfx1250, wave32).
//
//   K1: gather x = emb[seqs] (8 MB, L2-resident) + per-row L2 norms (float4).
//   K2: initial 2047 adjacent-pair dot products on the matrix engine
//       (V_WMMA_F32_16X16X4_F32, exact f32), b128 loads via column remap.
//   K3: single-workgroup persistent kernel: 2047 incremental merge steps,
//       O(S) argmax + O(E) compose per step, LDS-resident state,
//       workgroup-scope fences only. Latency-bound by design.
// ---------------------------------------------------------------------------

#define SLEN 2048
#define EDIM 1024
#define BANYAN_EPS 1e-8f

typedef __attribute__((ext_vector_type(2))) float v2f;
typedef __attribute__((ext_vector_type(8))) float v8f;

// ---- wave32 helpers --------------------------------------------------------

__device__ __forceinline__ float waveReduceSumF32(float v) {
    #pragma unroll
    for (int off = 16; off > 0; off >>= 1) v += __shfl_xor(v, off, 32);
    return v;
}

__device__ __forceinline__ unsigned long long waveReduceMaxU64(unsigned long long k) {
    #pragma unroll
    for (int off = 16; off > 0; off >>= 1) {
        unsigned long long o = __shfl_xor(k, off, 32);
        k = (o > k) ? o : k;
    }
    return k;
}

// Order-preserving (value-major, then lowest-index) key. Max key == argmax
// with jnp.argmax tie-breaking (first/lowest index wins among equal values).
__device__ __forceinline__ unsigned long long simKey(float s, int idx) {
    unsigned u = __float_as_uint(s);
    u = (u & 0x80000000u) ? ~u : (u | 0x80000000u);
    unsigned rid = (unsigned)((SLEN - 1) - idx);   // smaller idx -> larger rid
    return ((unsigned long long)u << 32) | (unsigned long long)rid;
}

// ---- K1: gather rows + norms (vectorized) ---------------------------------

__global__ void __launch_bounds__(256)
banyan_gather_norm(const int* __restrict__ seqs, const float* __restrict__ emb,
                   float* __restrict__ x, float* __restrict__ nrm) {
    __shared__ float wsum[8];
    const int row = blockIdx.x;
    const int tok = seqs[row];
    const float4* src = (const float4*)(emb + (size_t)tok * EDIM);
    float4*       dst = (float4*)(x + (size_t)row * EDIM);
    float4 v = src[threadIdx.x];            // 256 threads x 4 floats = 1024
    dst[threadIdx.x] = v;
    float ss = fmaf(v.x, v.x, fmaf(v.y, v.y, fmaf(v.z, v.z, v.w * v.w)));
    ss = waveReduceSumF32(ss);
    if ((threadIdx.x & 31) == 0) wsum[threadIdx.x >> 5] = ss;
    __syncthreads();
    if (threadIdx.x == 0) {
        float t = 0.f;
        #pragma unroll
        for (int w = 0; w < 8; ++w) t += wsum[w];
        nrm[row] = sqrtf(t);
    }
}

// ---- K2: initial pair sims on the matrix engine ---------------------------
// One wave per 16-pair tile; D[m][n] = dot(x[p0+m], x[p0+1+n]); keep diagonal.
// Column remap trick: per 8-column block, lower lanes load float4 @kb, upper
// lanes @kb+4. WMMA0 consumes columns {kb,kb+1 | kb+4,kb+5} (lower|upper K
// slots), WMMA1 consumes {kb+2,kb+3 | kb+6,kb+7}. Dot products are invariant
// under identical column permutation of A and B, so the diagonal is exact.
// D(16x16 f32) layout: lanes 0-15: (M=v, N=lane); lanes 16-31: (M=v+8, N=lane-16).

__global__ void __launch_bounds__(256)
banyan_init_sims_wmma(const float* __restrict__ x, const float* __restrict__ nrm,
                      float* __restrict__ sims) {
    const int wave = (blockIdx.x * 256 + threadIdx.x) >> 5;  // 0..127 tiles
    const int lane = threadIdx.x & 31;
    const int p0   = wave * 16;                 // first left index of tile
    const int row  = lane & 15;
    const int koff = (lane < 16) ? 0 : 4;
    const int li   = min(p0 + row, SLEN - 2);   // clamp partial tile (loads only)
    const float* L = x + (size_t)li * EDIM + koff;        // left rows  (A)
    const float* R = x + (size_t)(li + 1) * EDIM + koff;  // right rows (B)

    v8f acc = {};
    #pragma unroll 4
    for (int kb = 0; kb < EDIM; kb += 8) {
        float4 av = *(const float4*)(L + kb);
        float4 bv = *(const float4*)(R + kb);
        v2f a0, b0, a1, b1;
        a0.x = av.x; a0.y = av.y;  b0.x = bv.x; b0.y = bv.y;
        a1.x = av.z; a1.y = av.w;  b1.x = bv.z; b1.y = bv.w;
        // (neg_a, A, neg_b, B, c_mod, C, reuse_a, reuse_b)
        acc = __builtin_amdgcn_wmma_f32_16x16x4_f32(
            false, a0, false, b0, (short)0, acc, false, false);
        acc = __builtin_amdgcn_wmma_f32_16x16x4_f32(
            false, a1, false, b1, (short)0, acc, false, false);
    }

    // extract diagonal and finish cosine
    int m, vi;
    if (lane < 8)        { m = lane;      vi = lane; }
    else if (lane >= 24) { m = lane - 16; vi = lane - 24; }
    else                 { m = -1;        vi = 0; }
    float d = acc[vi];
    if (m >= 0) {
        int p = p0 + m;
        if (p < SLEN - 1) {
            float dl = fmaxf(nrm[p],     BANYAN_EPS);
            float dr = fmaxf(nrm[p + 1], BANYAN_EPS);
            sims[p] = d / (dl * dr);
        }
    }
}

// ---- K3: persistent incremental merge -------------------------------------

__global__ void __launch_bounds__(256)
banyan_merge(float* __restrict__ x, const float* __restrict__ nrmG,
             const float* __restrict__ simsG,
             const float* __restrict__ WlG, const float* __restrict__ WrG,
             const float* __restrict__ bG, float* __restrict__ out) {
    __shared__ __align__(16) float sims[SLEN];
    __shared__ __align__(16) float nrm[SLEN];
    __shared__ int   prv[SLEN];
    __shared__ int   nxt[SLEN];
    __shared__ __align__(16) float wl[EDIM], wr[EDIM], bb[EDIM];
    __shared__ unsigned long long bestKey;
    __shared__ float acc[3];   // [ssParent, dotLeft, dotRight]

    const int tid = threadIdx.x;

    for (int p = tid; p < SLEN; p += 256) {
        sims[p] = (p < SLEN - 1) ? simsG[p] : -INFINITY;
        nrm[p]  = nrmG[p];
        prv[p]  = p - 1;
        nxt[p]  = p + 1;
    }
    for (int e = tid; e < EDIM; e += 256) {
        wl[e] = WlG[e]; wr[e] = WrG[e]; bb[e] = bG[e];
    }
    __syncthreads();

    const int e0 = tid * 4;   // this thread's contiguous 4-element slice of E
    const float4 wl4 = *(const float4*)(wl + e0);
    const float4 wr4 = *(const float4*)(wr + e0);
    const float4 bb4 = *(const float4*)(bb + e0);

    for (int step = 0; step < SLEN - 1; ++step) {
        if (tid == 0) { bestKey = 0ull; acc[0] = acc[1] = acc[2] = 0.f; }
        __syncthreads();

        // --- argmax over sims: 2x float4 LDS reads per thread, key-encoded ---
        unsigned long long k = 0ull;
        {
            const float4* s4 = (const float4*)sims;
            #pragma unroll
            for (int q = 0; q < 2; ++q) {
                int base = tid * 8 + q * 4;
                float4 sv = s4[tid * 2 + q];
                unsigned long long k0 = simKey(sv.x, base + 0);
                unsigned long long k1 = simKey(sv.y, base + 1);
                unsigned long long k2 = simKey(sv.z, base + 2);
                unsigned long long k3 = simKey(sv.w, base + 3);
                k0 = (k1 > k0) ? k1 : k0;
                k2 = (k3 > k2) ? k3 : k2;
                k0 = (k2 > k0) ? k2 : k0;
                k  = (k0 > k)  ? k0 : k;
            }
        }
        k = waveReduceMaxU64(k);
        if ((tid & 31) == 0) atomicMax(&bestKey, k);   // ds_max_u64
        __syncthreads();

        const int i = (SLEN - 1) - (int)(bestKey & 0xFFFFFFFFull);
        const int j = nxt[i];
        const int p = prv[i];
        const int n = nxt[j];

        const float* xi = x + (size_t)i * EDIM;
        float*       xj = x + (size_t)j * EDIM;
        const float* xp = x + (size_t)((p >= 0)   ? p : i) * EDIM;  // guarded
        const float* xn = x + (size_t)((n < SLEN) ? n : i) * EDIM;  // guarded
        __builtin_prefetch(xp + e0, 0, 1);   // global_prefetch_b8
        __builtin_prefetch(xn + e0, 0, 1);

        // --- fused: parent = tanh(l*wl + r*wr + b); 3 reductions; b128 I/O ---
        float4 l4 = *(const float4*)(xi + e0);
        float4 r4 = *(const float4*)(xj + e0);
        float4 p4 = *(const float4*)(xp + e0);
        float4 n4 = *(const float4*)(xn + e0);
        float4 pv;
        pv.x = tanhf(fmaf(l4.x, wl4.x, fmaf(r4.x, wr4.x, bb4.x)));
        pv.y = tanhf(fmaf(l4.y, wl4.y, fmaf(r4.y, wr4.y, bb4.y)));
        pv.z = tanhf(fmaf(l4.z, wl4.z, fmaf(r4.z, wr4.z, bb4.z)));
        pv.w = tanhf(fmaf(l4.w, wl4.w, fmaf(r4.w, wr4.w, bb4.w)));
        *(float4*)(xj + e0) = pv;            // parent overwrites right slot

        float ss = fmaf(pv.x, pv.x, fmaf(pv.y, pv.y, fmaf(pv.z, pv.z, pv.w * pv.w)));
        float dl = fmaf(p4.x, pv.x, fmaf(p4.y, pv.y, fmaf(p4.z, pv.z, p4.w * pv.w)));
        float dr = fmaf(n4.x, pv.x, fmaf(n4.y, pv.y, fmaf(n4.z, pv.z, n4.w * pv.w)));

        ss = waveReduceSumF32(ss);
        dl = waveReduceSumF32(dl);
        dr = waveReduceSumF32(dr);
        if ((tid & 31) == 0) {
            atomicAdd(&acc[0], ss);          // ds_add_f32
            atomicAdd(&acc[1], dl);
            atomicAdd(&acc[2], dr);
        }
        __threadfence_block();  // workgroup-scope: xj stores visible to all waves
        __syncthreads();

        // --- scalar bookkeeping: only two sims change ---
        if (tid == 0) {
            float nj  = sqrtf(acc[0]);
            nrm[j]    = nj;
            float njc = fmaxf(nj, BANYAN_EPS);
            sims[i]   = -INFINITY;           // slot i removed
            if (p >= 0) {
                nxt[p] = j; prv[j] = p;
                sims[p] = acc[1] / (fmaxf(nrm[p], BANYAN_EPS) * njc);
            } else {
                prv[j] = -1;
            }
            sims[j] = (n < SLEN) ? (acc[2] / (njc * fmaxf(nrm[n], BANYAN_EPS)))
                                 : -INFINITY;
        }
        __syncthreads();
    }

    // survivor is always slot S-1 (a rightmost element is never a left index)
    const float4* root = (const float4*)(x + (size_t)(SLEN - 1) * EDIM);
    ((float4*)out)[tid] = root[tid];
}

// ---- host launcher ---------------------------------------------------------

extern "C" void kernel_launch(void* const* d_in, const int* in_sizes, int n_in,
                              void* d_out, int out_size, void* d_ws, size_t ws_size,
                              hipStream_t stream) {
    const int*   seqs = (const int*)  d_in[0];
    const float* emb  = (const float*)d_in[1];
    const float* Wl   = (const float*)d_in[2];   // [C,ED] == flat [E]
    const float* Wr   = (const float*)d_in[3];
    const float* b    = (const float*)d_in[4];
    float* out = (float*)d_out;

    float* x    = (float*)d_ws;                       // S*E
    float* nrm  = x + (size_t)SLEN * EDIM;            // S
    float* sims = nrm + SLEN;                         // S

    banyan_gather_norm<<<SLEN, 256, 0, stream>>>(seqs, emb, x, nrm);
    // 2047 pairs -> 128 tiles of 16; 8 waves/block -> 16 blocks exactly
    banyan_init_sims_wmma<<<16, 256, 0, stream>>>(x, nrm, sims);
    banyan_merge<<<1, 256, 0, stream>>>(x, nrm, sims, Wl, Wr, b, out);
}